// QuantumAttentionLayer_19224273617440
// MI455X (gfx1250) — compile-verified
//
#include <hip/hip_runtime.h>
#include <math.h>

// ---------------------------------------------------------------------------
// QuantumAttentionLayer, algebraically collapsed.
//
// Circuit: H^n -> CNOT chain -> per-qubit RX(data) RY(ry) RZ(rz), then <Z_i>.
//  * CNOT chain on the uniform state is the identity (permutation of a
//    constant vector): the state stays the product state |+>^n.
//  * RX on |+> is a global phase (|+> is an X eigenstate) -> the per-sample
//    data encoding (query/key/value) never affects probabilities.
//  * RZ is diagonal -> phases only, no effect on |amp|^2.
//  => <Z_i> = <Z> of RY(b)|+> = -sin(ry_theta[h,i]), identical for every
//     batch row. The layer reduces to:
//
//       out[b, j] = bias[j] - sum_k sin(ry_flat[k]) * W[j, k]   for all b.
//
// Remaining work: 20 sines + a 1x20 @ 20x20 projection (done exactly on the
// gfx1250 fp32 WMMA path, v_wmma_f32_16x16x4_f32) + a 328 KB broadcast store
// (HBM floor ~14 ns @ 23.3 TB/s -> pure launch latency).
// ---------------------------------------------------------------------------

typedef float v2f __attribute__((ext_vector_type(2)));
typedef float v8f __attribute__((ext_vector_type(8)));

#define NQ 20  // NQUBIT

__global__ void __launch_bounds__(256) qattn_collapsed_kernel(
    const float* __restrict__ ry,    // (NHEAD, HEAD_DIM) flat = 20 floats
    const float* __restrict__ W,     // (20, 20) row-major
    const float* __restrict__ bias,  // (20,)
    float* __restrict__ out,         // (rows, 20)
    int n4)                          // total float4 elements in out
{
  __shared__ __align__(16) float crow[32];

  const int tid = threadIdx.x;

  // ---- Phase 1: wave 0 computes crow[j] = bias[j] - sin(ry) . W[j,:] ------
  // fp32 WMMA: D(16x16) += A(16x4) * B(4x16), K=20 in 5 chunks of 4.
  //   A: row 0 = xrow = -sin(ry), rows 1..15 = 0  (ISA 32-bit 16x4 A layout:
  //      lane = M row, lane[4] selects K pair, VGPR = K parity).
  //   B[k][n] = W[n][k]  =>  D[0][n] = sum_k xrow[k] * W[n][k].
  if (tid < 32) {                    // exactly wave 0: EXEC all 1s inside
    const int lane = tid;
    const int m  = lane & 15;        // M row of the tile this lane holds
    const int hi = lane >> 4;        // 0 -> K pair {0,1}, 1 -> K pair {2,3}

    #pragma unroll
    for (int t = 0; t < 2; ++t) {            // N tiles: cols [0,16), [16,20)
      const int n = t * 16 + m;              // output column fed by this lane
      v8f acc = {};
      #pragma unroll
      for (int c = 0; c < 5; ++c) {
        const int kk = 4 * c + 2 * hi;       // lane's K-pair base (max 19)
        v2f a, b;
        // __sinf: native v_sin_f32 path; angles are O(1) trainable params,
        // avoids inlining the 2k-instruction Payne-Hanek slow path 20x.
        a.x = (m == 0) ? -__sinf(ry[kk])     : 0.0f;
        a.y = (m == 0) ? -__sinf(ry[kk + 1]) : 0.0f;
        b.x = (n < NQ) ? W[n * NQ + kk]     : 0.0f;
        b.y = (n < NQ) ? W[n * NQ + kk + 1] : 0.0f;
        acc = __builtin_amdgcn_wmma_f32_16x16x4_f32(
            /*neg_a=*/false, a, /*neg_b=*/false, b,
            /*c_mod=*/(short)0, acc, /*reuse_a=*/false, /*reuse_b=*/false);
      }
      // D row M=0 lives in element 0 (VGPR0) of lanes 0..15.
      if (lane < 16 && n < NQ) crow[n] = acc[0] + bias[n];
    }
  }
  __syncthreads();

  // ---- Phase 2: broadcast the constant row; 20 floats = 5 float4 / row ----
  // Flat float4 index idx covers out[row][4c..4c+3] with c = idx % 5.
  const float4* crowv = (const float4*)crow;
  float4* dst = (float4*)out;
  int idx = blockIdx.x * blockDim.x + tid;
  const int stride = gridDim.x * blockDim.x;
  for (; idx < n4; idx += stride) {
    dst[idx] = crowv[idx % 5];
  }
}

extern "C" void kernel_launch(void* const* d_in, const int* in_sizes, int n_in,
                              void* d_out, int out_size, void* d_ws, size_t ws_size,
                              hipStream_t stream) {
  // setup_inputs order: query(0), key(1), value(2), ry_theta(3), rz_theta(4),
  // W(5), b(6). query/key/value/rz_theta provably do not affect the output
  // (global phase / diagonal phase / permutation of a uniform state).
  const float* ry   = (const float*)d_in[3];
  const float* W    = (const float*)d_in[5];
  const float* bias = (const float*)d_in[6];
  float* out = (float*)d_out;

  const int n4 = out_size / 4;                 // 81920 floats -> 20480 float4
  int blocks = (n4 + 255) / 256;               // 80 blocks of 256 (8 waves ea.)
  if (blocks < 1) blocks = 1;
  qattn_collapsed_kernel<<<blocks, 256, 0, stream>>>(ry, W, bias, out, n4);
}